// GRN_6442450944303
// MI455X (gfx1250) — compile-verified
//
#include <hip/hip_runtime.h>
#include <hip/hip_bf16.h>

// ---- problem constants (match reference) ----
constexpr int B = 4, N = 512, E = 128, R = 32, C = 64, T = 9;
constexpr int D = E + R;          // 160
constexpr int NP = N + 4;         // padded rows (2 zero rows each side, max pad=2)
constexpr int ZSTR = D + 4;       // 164: LDS row stride for z tile (164%64=36 -> conflict-free)
constexpr int XSTR = C + 4;       // 68:  LDS row stride for x tile via TDM pad (68%64=4)

typedef __attribute__((ext_vector_type(2))) float v2f;
typedef __attribute__((ext_vector_type(8))) float v8f;
typedef __attribute__((ext_vector_type(4))) unsigned u32x4;
typedef __attribute__((ext_vector_type(8))) unsigned u32x8;

__device__ __forceinline__ v8f wmma_f32_k4(v2f a, v2f b, v8f c) {
    // D = A(16x4,f32) * B(4x16,f32) + C(16x16,f32)
    return __builtin_amdgcn_wmma_f32_16x16x4_f32(
        /*neg_a=*/false, a, /*neg_b=*/false, b,
        /*c_mod=*/(short)0, c, /*reuse_a=*/false, /*reuse_b=*/false);
}

// per-lane async copy: 16B global -> LDS (tracked by ASYNCcnt)
__device__ __forceinline__ void async_g2l_b128(unsigned lds_addr, int gbyte_off,
                                               const float* gbase) {
    asm volatile("global_load_async_to_lds_b128 %0, %1, %2"
                 :: "v"(lds_addr), "v"(gbyte_off), "s"(gbase)
                 : "memory");
}
__device__ __forceinline__ void wait_async0() {
    asm volatile("s_wait_asynccnt 0x0" ::: "memory");
}

// ---------------------------------------------------------------------------
// Kernel A: char branch. cc[b][r] = max_n ( conv3(emb[char_ids], repre_w) + b )
// ---------------------------------------------------------------------------
__global__ void k_char_conv_max(const int* __restrict__ cid,
                                const float* __restrict__ emb,
                                const float* __restrict__ rw,
                                const float* __restrict__ rb,
                                float* __restrict__ cc) {
    int b = blockIdx.x, r = blockIdx.y;
    const float* w = rw + (size_t)r * 3 * E;
    float bias = rb[r];
    float lmax = -3.4e38f;
    for (int n = threadIdx.x; n < N; n += blockDim.x) {
        float s = bias;
        #pragma unroll
        for (int t = 0; t < 3; ++t) {
            int nn = n + t - 1;
            if (nn < 0 || nn >= N) continue;            // zero padding
            const float* er = emb + (size_t)cid[b * N + nn] * E;
            const float* wr = w + t * E;
            for (int e = 0; e < E; ++e) s = fmaf(er[e], wr[e], s);
        }
        lmax = fmaxf(lmax, s);
    }
    __shared__ float red[128];
    red[threadIdx.x] = lmax;
    __syncthreads();
    for (int s = 64; s > 0; s >>= 1) {
        if (threadIdx.x < s) red[threadIdx.x] = fmaxf(red[threadIdx.x], red[threadIdx.x + s]);
        __syncthreads();
    }
    if (threadIdx.x == 0) cc[b * R + r] = red[0];
}

// ---------------------------------------------------------------------------
// Kernel B: build padded z: z_pad[b][m][d], m=n+2; rows 0,1,N+2,N+3 are zero.
// ---------------------------------------------------------------------------
__global__ void k_build_z(const int* __restrict__ input_ids,
                          const float* __restrict__ emb,
                          const float* __restrict__ cc,
                          float* __restrict__ zp) {
    int idx = blockIdx.x * blockDim.x + threadIdx.x;
    if (idx >= B * NP * D) return;
    int d = idx % D;
    int m = (idx / D) % NP;
    int b = idx / (D * NP);
    float v = 0.f;
    if (m >= 2 && m < N + 2) {
        int n = m - 2;
        if (d < R) {
            if (b == 0 && n < B) v = cc[n * R + d];    // zeros/view quirk
        } else {
            v = emb[(size_t)input_ids[b * N + n] * E + (d - R)];
        }
    }
    zp[idx] = v;
}

// ---------------------------------------------------------------------------
// Kernel C: context layer. A-tile (20 rows x D) staged into LDS once per block
// via GLOBAL_LOAD_ASYNC_TO_LDS_B128 (4-wave reuse), bank-conflict-free stride.
// Per tap: K=D matmul via v_wmma_f32_16x16x4_f32, 4 K-steps batched per wait.
// ---------------------------------------------------------------------------
__global__ void k_context_conv(const float* __restrict__ zp,
                               const float* __restrict__ w1, const float* __restrict__ b1,
                               const float* __restrict__ w3, const float* __restrict__ b3,
                               const float* __restrict__ w5, const float* __restrict__ b5,
                               float* __restrict__ x) {
    __shared__ float zt[20 * ZSTR];
    int tile = blockIdx.x;              // 0..127
    int b    = tile >> 5;
    int n0   = (tile & 31) << 4;
    int tid  = threadIdx.x;
    int lane = tid & 31;
    int wave = tid >> 5;

    // ---- async stage: 20 rows x 160 f32 (contiguous in global) -> padded LDS
    const float* gbase = zp + (size_t)b * NP * D + (size_t)n0 * D;
    unsigned ldsbase = (unsigned)(unsigned long long)(&zt[0]);
    #pragma unroll
    for (int it = 0; it < 7; ++it) {
        int q = it * 128 + tid;                     // b128 chunk, 40 per row
        if (q < 20 * 40) {
            int row = q / 40, cq = q - row * 40;
            unsigned la = ldsbase + (unsigned)(row * ZSTR + cq * 4) * 4u;
            async_g2l_b128(la, q * 16, gbase);
        }
    }
    wait_async0();
    __syncthreads();

    int c0  = wave << 4;
    int M   = lane & 15;                // A row / B column selector
    int sel = (lane >> 4) << 1;         // K sub-pair: lanes<16 -> {0,1}, >=16 -> {2,3}

    auto tap = [&](const float* lds_row, const float* wr, v8f acc) -> v8f {
        #pragma unroll
        for (int kb = 0; kb < D; kb += 16) {        // 4 WMMAs per wait-group
            v2f a0 = *(const v2f*)(lds_row + kb);
            v2f a1 = *(const v2f*)(lds_row + kb + 4);
            v2f a2 = *(const v2f*)(lds_row + kb + 8);
            v2f a3 = *(const v2f*)(lds_row + kb + 12);
            v2f f0 = *(const v2f*)(wr + kb);
            v2f f1 = *(const v2f*)(wr + kb + 4);
            v2f f2 = *(const v2f*)(wr + kb + 8);
            v2f f3 = *(const v2f*)(wr + kb + 12);
            acc = wmma_f32_k4(a0, f0, acc);
            acc = wmma_f32_k4(a1, f1, acc);
            acc = wmma_f32_k4(a2, f2, acc);
            acc = wmma_f32_k4(a3, f3, acc);
        }
        return acc;
    };

    v8f y1 = {}; v8f y3 = {}; v8f y5 = {};
    // local padded row for output n0+M, tap t, pad p: (M + 2 - p + t)
    y1 = tap(zt + (M + 2) * ZSTR + sel, w1 + (size_t)(c0 + M) * D + sel, y1);
    #pragma unroll
    for (int t = 0; t < 3; ++t)
        y3 = tap(zt + (M + 1 + t) * ZSTR + sel,
                 w3 + (size_t)(c0 + M) * (3 * D) + t * D + sel, y3);
    #pragma unroll
    for (int t = 0; t < 5; ++t)
        y5 = tap(zt + (M + t) * ZSTR + sel,
                 w5 + (size_t)(c0 + M) * (5 * D) + t * D + sel, y5);

    int col   = lane & 15;
    int mbase = (lane >> 4) << 3;       // D-frag: M = v + 8*(lane>=16)
    float bi1 = b1[c0 + col], bi3 = b3[c0 + col], bi5 = b5[c0 + col];
    #pragma unroll
    for (int v = 0; v < 8; ++v) {
        int row = n0 + mbase + v;
        float val = fmaxf(fmaxf(tanhf(y1[v] + bi1), tanhf(y3[v] + bi3)),
                          tanhf(y5[v] + bi5));
        x[((size_t)b * N + row) * C + c0 + col] = val;
    }
}

// ---------------------------------------------------------------------------
// Kernel D: g0/g1 = x@fgX_w.T + bX. The 16x64 x-tile is staged by the Tensor
// Data Mover (tensor_load_to_lds, wave 0 only, TENSORcnt) with pad_enable
// (interval 64 dw, amount 4 dw) -> LDS stride 68, bank-conflict-free.
// ---------------------------------------------------------------------------
__global__ void k_gates(const float* __restrict__ x,
                        const float* __restrict__ w0, const float* __restrict__ bb0,
                        const float* __restrict__ w1, const float* __restrict__ bb1,
                        float* __restrict__ g0, float* __restrict__ g1) {
    __shared__ float xt[16 * XSTR];
    int m0   = blockIdx.x << 4;         // global row in (B*N)
    int tid  = threadIdx.x;
    int lane = tid & 31;
    int wave = tid >> 5;

    if (wave == 0) {                    // TDM is per-wave (EXEC ignored)
        unsigned long long ga = (unsigned long long)(x + (size_t)m0 * C);
        unsigned la = (unsigned)(unsigned long long)(&xt[0]);
        u32x4 d0; u32x8 d1;
        d0[0] = 1u;                                         // count=1, load, no gather
        d0[1] = la;                                         // lds_addr
        d0[2] = (unsigned)ga;                               // global_addr[31:0]
        d0[3] = (unsigned)(ga >> 32) | (2u << 30);          // addr[56:32] | type=2
        d1[0] = (2u << 16)                                  // data_size = 4B
              | (1u << 20)                                  // pad_enable
              | (5u << 22)                                  // pad_interval = 64 dwords
              | (3u << 25);                                 // pad_amount  = 4 dwords
        d1[1] = (unsigned)C << 16;                          // tensor_dim0 = 64
        d1[2] = 16u << 16;                                  // tensor_dim1 = 16
        d1[3] = (unsigned)C << 16;                          // tile_dim0 = 64
        d1[4] = 16u;                                        // tile_dim1 = 16
        d1[5] = (unsigned)C;                                // tensor_dim0_stride = 64
        d1[6] = 0u;
        d1[7] = 0u;
        asm volatile("tensor_load_to_lds %0, %1" :: "s"(d0), "s"(d1) : "memory");
        __builtin_amdgcn_s_wait_tensorcnt(0);
    }
    __syncthreads();

    int c0  = wave << 4;
    int M   = lane & 15;
    int sel = (lane >> 4) << 1;
    const float* ar  = xt + M * XSTR + sel;
    const float* w0r = w0 + (size_t)(c0 + M) * C + sel;
    const float* w1r = w1 + (size_t)(c0 + M) * C + sel;
    v8f a0 = {}; v8f a1 = {};
    #pragma unroll
    for (int kb = 0; kb < C; kb += 16) {
        v2f av0 = *(const v2f*)(ar + kb);
        v2f av1 = *(const v2f*)(ar + kb + 4);
        v2f av2 = *(const v2f*)(ar + kb + 8);
        v2f av3 = *(const v2f*)(ar + kb + 12);
        v2f f00 = *(const v2f*)(w0r + kb);
        v2f f01 = *(const v2f*)(w0r + kb + 4);
        v2f f02 = *(const v2f*)(w0r + kb + 8);
        v2f f03 = *(const v2f*)(w0r + kb + 12);
        v2f f10 = *(const v2f*)(w1r + kb);
        v2f f11 = *(const v2f*)(w1r + kb + 4);
        v2f f12 = *(const v2f*)(w1r + kb + 8);
        v2f f13 = *(const v2f*)(w1r + kb + 12);
        a0 = wmma_f32_k4(av0, f00, a0);
        a0 = wmma_f32_k4(av1, f01, a0);
        a0 = wmma_f32_k4(av2, f02, a0);
        a0 = wmma_f32_k4(av3, f03, a0);
        a1 = wmma_f32_k4(av0, f10, a1);
        a1 = wmma_f32_k4(av1, f11, a1);
        a1 = wmma_f32_k4(av2, f12, a1);
        a1 = wmma_f32_k4(av3, f13, a1);
    }
    int col = lane & 15;
    int mbase = (lane >> 4) << 3;
    float bi0 = bb0[c0 + col], bi1 = bb1[c0 + col];
    #pragma unroll
    for (int v = 0; v < 8; ++v) {
        size_t o = (size_t)(m0 + mbase + v) * C + c0 + col;
        g0[o] = a0[v] + bi0;
        g1[o] = a1[v] + bi1;
    }
}

// ---------------------------------------------------------------------------
// Kernel E: relation layer + classifier, one block per (b,i), 256 threads.
// ---------------------------------------------------------------------------
__global__ void k_relation(const float* __restrict__ x,
                           const float* __restrict__ g0,
                           const float* __restrict__ g1,
                           const int* __restrict__ amask,
                           const float* __restrict__ clsw,
                           const float* __restrict__ clsb,
                           float* __restrict__ out) {
    int b = blockIdx.x >> 9;            // / N
    int i = blockIdx.x & (N - 1);
    int tid = threadIdx.x;
    __shared__ float g0row[C];
    __shared__ float part[4][C];
    __shared__ float rbuf[C];
    __shared__ float lenred[256];
    __shared__ float mrow[N];

    float ls = 0.f;
    for (int j = tid; j < N; j += 256) {
        float mv = (float)amask[b * N + j];
        mrow[j] = mv;
        ls += mv;
    }
    lenred[tid] = ls;
    __syncthreads();
    for (int s = 128; s > 0; s >>= 1) {
        if (tid < s) lenred[tid] += lenred[tid + s];
        __syncthreads();
    }
    float len = lenred[0];

    if (tid < C) g0row[tid] = g0[((size_t)b * N + i) * C + tid];
    __syncthreads();

    int c = tid & (C - 1), js = tid >> 6;
    float acc = 0.f;
    for (int j = js; j < N; j += 4) {
        float t  = g0row[c] + g1[((size_t)b * N + j) * C + c];
        float gate = 1.f / (1.f + __expf(-t));
        acc = fmaf(x[((size_t)b * N + j) * C + c] * mrow[j], gate, acc);
    }
    part[js][c] = acc;
    __syncthreads();
    if (tid < C) {
        float s = (part[0][tid] + part[1][tid] + part[2][tid] + part[3][tid]) * mrow[i];
        rbuf[tid] = tanhf(s / len);
    }
    __syncthreads();
    if (tid < T) {
        float s = clsb[tid];
        for (int cc2 = 0; cc2 < C; ++cc2) s = fmaf(rbuf[cc2], clsw[tid * C + cc2], s);
        out[((size_t)b * N + i) * T + tid] = s;
    }
}

// ---------------------------------------------------------------------------
// Kernel F: torchcrf-style NLL (sum). One block, 64 threads.
// ---------------------------------------------------------------------------
__global__ void k_crf(const float* __restrict__ em,
                      const int* __restrict__ labels,
                      const int* __restrict__ amask,
                      const float* __restrict__ start,
                      const float* __restrict__ endw,
                      const float* __restrict__ trans,
                      float* __restrict__ out) {
    __shared__ float tr[T * T], st[T], en[T];
    __shared__ float alpha[B][T];
    __shared__ float scoreL[B], logZ[B];
    int tid = threadIdx.x;
    if (tid < T * T) tr[tid] = trans[tid];
    if (tid < T) { st[tid] = start[tid]; en[tid] = endw[tid]; }
    __syncthreads();

    if (tid < B) {                       // gold-path numerator (no barriers)
        int bb = tid;
        int prev = labels[bb * N];
        float sc = st[prev] + em[((size_t)bb * N) * T + prev];
        for (int n = 1; n < N; ++n) {
            int tg = labels[bb * N + n];
            float m = (float)amask[bb * N + n];
            sc += (tr[prev * T + tg] + em[((size_t)bb * N + n) * T + tg]) * m;
            if (m > 0.f) prev = tg;
        }
        scoreL[bb] = sc + en[prev];
    }
    if (tid < B * T) {
        int bb = tid / T, t = tid % T;
        alpha[bb][t] = st[t] + em[((size_t)bb * N) * T + t];
    }
    __syncthreads();

    for (int n = 1; n < N; ++n) {        // forward algorithm
        float v = 0.f, keep = 0.f;
        int bb = 0, t = 0;
        if (tid < B * T) {
            bb = tid / T; t = tid % T;
            float mx = -3.4e38f;
            for (int k = 0; k < T; ++k) mx = fmaxf(mx, alpha[bb][k] + tr[k * T + t]);
            float s = 0.f;
            for (int k = 0; k < T; ++k) s += __expf(alpha[bb][k] + tr[k * T + t] - mx);
            v = mx + __logf(s) + em[((size_t)bb * N + n) * T + t];
            keep = alpha[bb][t];
        }
        __syncthreads();                 // all reads of alpha done
        if (tid < B * T) {
            float m = (float)amask[bb * N + n];
            alpha[bb][t] = (m > 0.f) ? v : keep;
        }
        __syncthreads();
    }
    if (tid < B) {
        int bb = tid;
        float mx = -3.4e38f;
        for (int t = 0; t < T; ++t) mx = fmaxf(mx, alpha[bb][t] + en[t]);
        float s = 0.f;
        for (int t = 0; t < T; ++t) s += __expf(alpha[bb][t] + en[t] - mx);
        logZ[bb] = mx + __logf(s);
    }
    __syncthreads();
    if (tid == 0) {
        float loss = 0.f;
        for (int bb = 0; bb < B; ++bb) loss += scoreL[bb] - logZ[bb];
        out[B * N * T] = -loss;          // d_out = [logits (B*N*T), loss]
    }
}

// ---------------------------------------------------------------------------
extern "C" void kernel_launch(void* const* d_in, const int* in_sizes, int n_in,
                              void* d_out, int out_size, void* d_ws, size_t ws_size,
                              hipStream_t stream) {
    const int*   input_ids = (const int*)d_in[0];
    const int*   labels    = (const int*)d_in[1];
    const int*   amask     = (const int*)d_in[2];
    const int*   char_ids  = (const int*)d_in[3];
    const float* emb       = (const float*)d_in[7];
    const float* repre_w   = (const float*)d_in[8];
    const float* repre_b   = (const float*)d_in[9];
    const float* c1_w      = (const float*)d_in[10];
    const float* c1_b      = (const float*)d_in[11];
    const float* c3_w      = (const float*)d_in[12];
    const float* c3_b      = (const float*)d_in[13];
    const float* c5_w      = (const float*)d_in[14];
    const float* c5_b      = (const float*)d_in[15];
    const float* fg0_w     = (const float*)d_in[16];
    const float* fg0_b     = (const float*)d_in[17];
    const float* fg1_w     = (const float*)d_in[18];
    const float* fg1_b     = (const float*)d_in[19];
    const float* cls_w     = (const float*)d_in[20];
    const float* cls_b     = (const float*)d_in[21];
    const float* crf_start = (const float*)d_in[22];
    const float* crf_end   = (const float*)d_in[23];
    const float* crf_trans = (const float*)d_in[24];
    float* out = (float*)d_out;

    // workspace carve-out (floats): cc | z_pad | x | g0 | g1  (~2.9 MB total)
    float* ws = (float*)d_ws;
    float* cc = ws;
    float* zp = cc + B * R;
    float* x  = zp + (size_t)B * NP * D;
    float* g0 = x  + (size_t)B * N * C;
    float* g1 = g0 + (size_t)B * N * C;

    k_char_conv_max<<<dim3(B, R), 128, 0, stream>>>(char_ids, emb, repre_w, repre_b, cc);
    k_build_z<<<(B * NP * D + 255) / 256, 256, 0, stream>>>(input_ids, emb, cc, zp);
    k_context_conv<<<(B * N) / 16, 128, 0, stream>>>(zp, c1_w, c1_b, c3_w, c3_b, c5_w, c5_b, x);
    k_gates<<<(B * N) / 16, 128, 0, stream>>>(x, fg0_w, fg0_b, fg1_w, fg1_b, g0, g1);
    k_relation<<<B * N, 256, 0, stream>>>(x, g0, g1, amask, cls_w, cls_b, out);
    k_crf<<<1, 64, 0, stream>>>(out, labels, amask, crf_start, crf_end, crf_trans, out);
}